// GCN_4183298146663
// MI455X (gfx1250) — compile-verified
//
#include <hip/hip_runtime.h>

#define N_NODES  100000
#define N_EDGES  3200000
#define N_GRAPHS 256
#define K_MAX    256   // largest GEMM K (head layers)

typedef float v2f __attribute__((ext_vector_type(2)));
typedef float v8f __attribute__((ext_vector_type(8)));

// ---------------------------------------------------------------------------
// Utility kernels
// ---------------------------------------------------------------------------
__global__ void zero_int_kernel(int* __restrict__ p, int n) {
  int i = blockIdx.x * blockDim.x + threadIdx.x;
  if (i < n) p[i] = 0;
}

__global__ void deg_count_kernel(const int* __restrict__ dst, int* __restrict__ cnt, int nE) {
  int e = blockIdx.x * blockDim.x + threadIdx.x;
  if (e < nE) atomicAdd(&cnt[dst[e]], 1);
}

__global__ void dinv_kernel(const int* __restrict__ cnt, float* __restrict__ dinv, int n) {
  int i = blockIdx.x * blockDim.x + threadIdx.x;
  if (i < n) dinv[i] = rsqrtf((float)(cnt[i] + 1));  // +1 for the self loop
}

// ---------------------------------------------------------------------------
// WMMA f32 GEMM: Y[MxN] = A[MxK] * B[KxN] (+ bias) (+ relu)
//  - grid.y = N/16  (one N-tile per block; its Kx16 B-strip staged in LDS once)
//  - grid.x covers M/16 in chunks of 4; each of the 4 waves does one 16x16 tile
//  - A fragment = one aligned 8-byte load (v2f of the consecutive K pair)
//  - B fragment served from LDS (ds_load), reused by all M-tiles of the block
// Requires M%16==0, N%16==0, K%4==0, K<=K_MAX (true for all GEMMs here).
// ---------------------------------------------------------------------------
template <int BIAS, int RELU>
__global__ void wmma_gemm_f32_kernel(const float* __restrict__ A, const float* __restrict__ B,
                                     const float* __restrict__ bias, float* __restrict__ Y,
                                     int M, int N, int K) {
  __shared__ float Bs[K_MAX * 16];  // 16 KB of the 320 KB WGP LDS

  const int ntile = blockIdx.y;
  const int lane  = threadIdx.x & 31;
  const int wave  = threadIdx.x >> 5;

  // ---- cooperative stage of B[K x 16] strip into LDS (float4 granularity) ----
  {
    const float* bcol = B + (size_t)ntile * 16;
    const int ngroups = K << 2;  // K rows * 4 float4-groups per row
    for (int g = threadIdx.x; g < ngroups; g += blockDim.x) {
      int k  = g >> 2;
      int c4 = (g & 3) << 2;
      float4 v = *(const float4*)(bcol + (size_t)k * N + c4);
      *(float4*)(&Bs[k * 16 + c4]) = v;
    }
  }
  __syncthreads();

  const int mtile = blockIdx.x * 4 + wave;
  if (mtile * 16 >= M) return;  // wave-uniform after the barrier: EXEC stays all-1s

  const int m15 = lane & 15;
  const int row = mtile * 16 + m15;   // A fragment row (M = lane&15)
  const int col = ntile * 16 + m15;   // B/C/D column  (N = lane&15)
  const int kh  = (lane >> 4) << 1;   // lanes 0-15 -> K 0,1 ; lanes 16-31 -> K 2,3

  v8f acc = {};
  const float* arow = A + (size_t)row * K;
  for (int k = 0; k < K; k += 4) {
    v2f a = *(const v2f*)(arow + k + kh);     // single global_load_b64
    v2f b;
    b.x = Bs[(k + kh) * 16 + m15];            // ds_load
    b.y = Bs[(k + kh + 1) * 16 + m15];
    acc = __builtin_amdgcn_wmma_f32_16x16x4_f32(
        /*neg_a=*/false, a, /*neg_b=*/false, b,
        /*c_mod=*/(short)0, acc, /*reuse_a=*/false, /*reuse_b=*/false);
  }

  const float bv = BIAS ? bias[col] : 0.0f;
  const int rbase = mtile * 16 + ((lane >> 4) << 3);  // VGPR j: rows j / j+8
#pragma unroll
  for (int j = 0; j < 8; ++j) {
    float v = acc[j] + bv;
    if (RELU) v = fmaxf(v, 0.0f);
    Y[(size_t)(rbase + j) * N + col] = v;
  }
}

// ---------------------------------------------------------------------------
// Self-loop init: Out[i][:] = Y[i][:] * dinv[i]^2   (replaces a zeroing pass)
// ---------------------------------------------------------------------------
__global__ void selfloop_init_kernel(const float* __restrict__ Y, const float* __restrict__ dinv,
                                     float* __restrict__ Out, int F4, int total) {
  int idx = blockIdx.x * blockDim.x + threadIdx.x;  // over n * F/4
  if (idx >= total) return;
  int i  = idx / F4;
  int fg = idx - i * F4;
  float w = dinv[i];
  w *= w;
  float4 v = ((const float4*)(Y + (size_t)i * (F4 << 2)))[fg];
  ((float4*)(Out + (size_t)i * (F4 << 2)))[fg] = make_float4(v.x * w, v.y * w, v.z * w, v.w * w);
}

// ---------------------------------------------------------------------------
// Edge scatter: Out[dst] += Y[src] * dinv[src]*dinv[dst]
// One thread per (edge, float4 group): a wave reads a contiguous slice of one
// (or a few) Y rows coalesced; atomics land in the L2-resident Out buffer.
// ---------------------------------------------------------------------------
__global__ void edge_scatter_kernel(const int* __restrict__ src, const int* __restrict__ dst,
                                    const float* __restrict__ dinv, const float* __restrict__ Y,
                                    float* __restrict__ Out, int F4, int nwork) {
  int idx = blockIdx.x * blockDim.x + threadIdx.x;
  if (idx >= nwork) return;
  int e  = idx / F4;
  int fg = idx - e * F4;
  int s = src[e];
  int d = dst[e];
  float w = dinv[s] * dinv[d];
  float4 v = ((const float4*)(Y + (size_t)s * (F4 << 2)))[fg];
  float* o = Out + (size_t)d * (F4 << 2) + (fg << 2);
  atomicAdd(o + 0, v.x * w);
  atomicAdd(o + 1, v.y * w);
  atomicAdd(o + 2, v.z * w);
  atomicAdd(o + 3, v.w * w);
}

// ---------------------------------------------------------------------------
// Fused bias + relu (in place). F is a power of two.
// ---------------------------------------------------------------------------
__global__ void bias_relu_kernel(float* __restrict__ H, const float* __restrict__ b,
                                 int Fmask, int total) {
  int idx = blockIdx.x * blockDim.x + threadIdx.x;
  if (idx >= total) return;
  int f = idx & Fmask;
  H[idx] = fmaxf(H[idx] + b[f], 0.0f);
}

// ---------------------------------------------------------------------------
// Pooling (segment max via sortable-uint atomicMax + segment mean via atomicAdd)
// ---------------------------------------------------------------------------
__device__ __forceinline__ unsigned float_to_sortable(float f) {
  unsigned b = __float_as_uint(f);
  return (b & 0x80000000u) ? ~b : (b | 0x80000000u);
}
__device__ __forceinline__ float sortable_to_float(unsigned u) {
  return (u & 0x80000000u) ? __uint_as_float(u ^ 0x80000000u) : __uint_as_float(~u);
}

__global__ void pool_init_kernel(unsigned* __restrict__ gmaxb, float* __restrict__ gsum,
                                 int* __restrict__ gcnt) {
  int i = blockIdx.x * blockDim.x + threadIdx.x;
  if (i < N_GRAPHS * 128) {
    gmaxb[i] = 0x007FFFFFu;  // encode(-inf)
    gsum[i]  = 0.0f;
  }
  if (i < N_GRAPHS) gcnt[i] = 0;
}

__global__ void pool_scatter_kernel(const float* __restrict__ H, const int* __restrict__ batch,
                                    unsigned* __restrict__ gmaxb, float* __restrict__ gsum,
                                    int* __restrict__ gcnt, int total) {
  int idx = blockIdx.x * blockDim.x + threadIdx.x;  // over N_NODES*128
  if (idx >= total) return;
  int i = idx >> 7;
  int f = idx & 127;
  int g = batch[i];
  float v = H[idx];
  atomicAdd(&gsum[(g << 7) + f], v);
  atomicMax(&gmaxb[(g << 7) + f], float_to_sortable(v));
  if (f == 0) atomicAdd(&gcnt[g], 1);
}

__global__ void pool_finalize_kernel(const unsigned* __restrict__ gmaxb,
                                     const float* __restrict__ gsum,
                                     const int* __restrict__ gcnt,
                                     float* __restrict__ pooled) {
  int idx = blockIdx.x * blockDim.x + threadIdx.x;  // over N_GRAPHS*128
  if (idx >= N_GRAPHS * 128) return;
  int g = idx >> 7;
  int f = idx & 127;
  float c = (float)max(gcnt[g], 1);
  pooled[(size_t)g * 256 + f]       = sortable_to_float(gmaxb[idx]);
  pooled[(size_t)g * 256 + 128 + f] = gsum[idx] / c;
}

// ---------------------------------------------------------------------------
// Host-side launch
// ---------------------------------------------------------------------------
static inline int nblk(long long n, int t) { return (int)((n + t - 1) / t); }

static void launch_gemm(const float* A, const float* B, const float* bias, float* Y,
                        int M, int N, int K, bool withBias, hipStream_t s) {
  dim3 grid(nblk(M / 16, 4), N / 16, 1);
  dim3 block(128, 1, 1);  // 4 waves, one 16x16 tile each
  if (withBias)
    wmma_gemm_f32_kernel<1, 0><<<grid, block, 0, s>>>(A, B, bias, Y, M, N, K);
  else
    wmma_gemm_f32_kernel<0, 0><<<grid, block, 0, s>>>(A, B, bias, Y, M, N, K);
}

extern "C" void kernel_launch(void* const* d_in, const int* in_sizes, int n_in,
                              void* d_out, int out_size, void* d_ws, size_t ws_size,
                              hipStream_t stream) {
  (void)in_sizes; (void)n_in; (void)out_size; (void)ws_size;

  const float* x     = (const float*)d_in[0];
  const int*   ei    = (const int*)d_in[1];
  const int*   src   = ei;
  const int*   dst   = ei + N_EDGES;
  const int*   batch = (const int*)d_in[2];

  const float* Ws[5] = {(const float*)d_in[3], (const float*)d_in[5], (const float*)d_in[7],
                        (const float*)d_in[9], (const float*)d_in[11]};
  const float* bs[5] = {(const float*)d_in[4], (const float*)d_in[6], (const float*)d_in[8],
                        (const float*)d_in[10], (const float*)d_in[12]};
  const int Ks[5] = {128, 16, 32, 32, 64};
  const int Fs[5] = {16, 32, 32, 64, 128};

  const float* Wbp  = (const float*)d_in[13];
  const float* bbp  = (const float*)d_in[14];
  const float* Wap  = (const float*)d_in[15];
  const float* bap  = (const float*)d_in[16];
  const float* Wout = (const float*)d_in[17];
  const float* bout = (const float*)d_in[18];

  // Workspace carve
  char* w = (char*)d_ws;
  auto carve = [&](size_t bytes) -> void* {
    void* p = (void*)w;
    w += (bytes + 255) & ~(size_t)255;
    return p;
  };
  int*      cnt    = (int*)carve((size_t)N_NODES * 4);
  float*    dinv   = (float*)carve((size_t)N_NODES * 4);
  float*    bufY   = (float*)carve((size_t)N_NODES * 128 * 4);
  float*    bufA   = (float*)carve((size_t)N_NODES * 128 * 4);
  float*    bufB   = (float*)carve((size_t)N_NODES * 128 * 4);
  unsigned* gmaxb  = (unsigned*)carve((size_t)N_GRAPHS * 128 * 4);
  float*    gsum   = (float*)carve((size_t)N_GRAPHS * 128 * 4);
  int*      gcnt   = (int*)carve((size_t)N_GRAPHS * 4);
  float*    pooled = (float*)carve((size_t)N_GRAPHS * 256 * 4);
  float*    hidden = (float*)carve((size_t)N_GRAPHS * 256 * 4);

  const int T = 256;

  // Degree / symmetric norm (shared by all conv layers)
  zero_int_kernel<<<nblk(N_NODES, T), T, 0, stream>>>(cnt, N_NODES);
  deg_count_kernel<<<nblk(N_EDGES, T), T, 0, stream>>>(dst, cnt, N_EDGES);
  dinv_kernel<<<nblk(N_NODES, T), T, 0, stream>>>(cnt, dinv, N_NODES);

  // 5 GCN conv layers
  const float* Hin = x;
  float* pp[2] = {bufA, bufB};
  int flip = 0;
  for (int l = 0; l < 5; ++l) {
    int K = Ks[l], F = Fs[l], F4 = F >> 2;
    // 1) projection: Y = Hin @ W   (WMMA f32, B staged in LDS)
    launch_gemm(Hin, Ws[l], nullptr, bufY, N_NODES, F, K, false, stream);
    // 2) self-loop term initializes the aggregation buffer
    float* Hout = pp[flip];
    int tot_nf4 = N_NODES * F4;
    selfloop_init_kernel<<<nblk(tot_nf4, T), T, 0, stream>>>(bufY, dinv, Hout, F4, tot_nf4);
    // 3) edge scatter-add (atomics, L2-resident destination)
    long long nwork = (long long)N_EDGES * F4;
    edge_scatter_kernel<<<nblk(nwork, T), T, 0, stream>>>(src, dst, dinv, bufY, Hout,
                                                          F4, (int)nwork);
    // 4) bias + relu (double relu in layer 1 is idempotent)
    int tot_nf = N_NODES * F;
    bias_relu_kernel<<<nblk(tot_nf, T), T, 0, stream>>>(Hout, bs[l], F - 1, tot_nf);
    Hin = Hout;
    flip ^= 1;
  }

  // lin_before_pool: Hbp = H @ Wbp + bbp (no relu)
  float* Hbp = pp[flip];
  launch_gemm(Hin, Wbp, bbp, Hbp, N_NODES, 128, 128, true, stream);

  // global max + mean pool per graph
  pool_init_kernel<<<nblk(N_GRAPHS * 128, T), T, 0, stream>>>(gmaxb, gsum, gcnt);
  pool_scatter_kernel<<<nblk((long long)N_NODES * 128, T), T, 0, stream>>>(
      Hbp, batch, gmaxb, gsum, gcnt, N_NODES * 128);
  pool_finalize_kernel<<<nblk(N_GRAPHS * 128, T), T, 0, stream>>>(gmaxb, gsum, gcnt, pooled);

  // head: hidden = pooled @ Wap + bap ; out = hidden @ Wout + bout
  launch_gemm(pooled, Wap, bap, hidden, N_GRAPHS, 256, 256, true, stream);
  launch_gemm(hidden, Wout, bout, (float*)d_out, N_GRAPHS, 128, 256, true, stream);
}